// RandGCONV_15109694947759
// MI455X (gfx1250) — compile-verified
//
#include <hip/hip_runtime.h>

// ---------------- problem constants (match reference) ----------------
#define BB      64          // batch
#define NN      4096        // nodes
#define DI      64
#define DH      64
#define DD      128         // DI+DH
#define SSUP    2
#define EE      131072      // edges per support
#define DOUT    128
#define NUM_MAT 5
#define ROWLEN  (BB*DD)     // 8192 floats per node-row, layout [n][b][d]
#define XMAT_ELEMS ((size_t)NN * ROWLEN)   // 33,554,432 floats = 128 MB

// W panel in LDS: pair-interleaved so a B fragment is one aligned ds_load_b64.
// element (d, j) lives at (d>>1)*PSTRIDE + 2*j + (d&1).
// PSTRIDE = 2*DOUT + 32 = 288 dwords: 288 % 64 == 32, so the half-wave reading
// pair-row p and the half-wave reading p+1 hit disjoint 32-bank halves.
#define PSTRIDE 288
#define PANEL   (64 * PSTRIDE)      // 18432 dwords = 72 KB per panel

typedef float v2f __attribute__((ext_vector_type(2)));
typedef float v8f __attribute__((ext_vector_type(8)));

// ---------------- pack: x0[n][b][d] = concat(inputs, state) ----------------
__global__ __launch_bounds__(256) void pack_x0(const float* __restrict__ inp,
                                               const float* __restrict__ st,
                                               float* __restrict__ x0) {
    size_t t = (size_t)blockIdx.x * 256 + threadIdx.x;     // over N*B*D
    int d = (int)(t & (DD - 1));
    int b = (int)((t >> 7) & (BB - 1));
    int n = (int)(t >> 13);
    float v = (d < DI) ? inp[((size_t)b * NN + n) * DI + d]
                       : st[((size_t)b * NN + n) * DH + (d - DI)];
    x0[t] = v;
}

// ---------------- zero a 128MB buffer (float4 stores) ----------------
__global__ __launch_bounds__(256) void zero_buf(float4* __restrict__ p) {
    size_t t = (size_t)blockIdx.x * 256 + threadIdx.x;
    p[t] = make_float4(0.f, 0.f, 0.f, 0.f);
}

// ---------------- COO scatter SpMM: y[row] += val * h[col] ----------------
// one block per edge; 256 threads x float4 x 8 iters cover the 8192-row.
// atomics resolve in L2 (h and y are L2-resident: 128MB each vs 192MB L2).
__global__ __launch_bounds__(256) void spmm_scatter(const int* __restrict__ rows,
                                                    const int* __restrict__ cols,
                                                    const float* __restrict__ vals,
                                                    const float* __restrict__ h,
                                                    float* __restrict__ y) {
    int e = blockIdx.x;
    int row = rows[e];
    int col = cols[e];
    float v = vals[e];
    const float* hp = h + (size_t)col * ROWLEN;
    float* yp = y + (size_t)row * ROWLEN;
    for (int i = threadIdx.x * 4; i < ROWLEN; i += 256 * 4) {
        float4 hv = *(const float4*)(hp + i);
        atomicAdd(yp + i + 0, v * hv.x);
        atomicAdd(yp + i + 1, v * hv.y);
        atomicAdd(yp + i + 2, v * hv.z);
        atomicAdd(yp + i + 3, v * hv.w);
    }
}

// ---------------- Chebyshev combine: x2 = 2*x2 - x0 (in place) ----------------
__global__ __launch_bounds__(256) void cheb_axpy(float4* __restrict__ x2,
                                                 const float4* __restrict__ x0) {
    size_t t = (size_t)blockIdx.x * 256 + threadIdx.x;
    float4 a = x2[t];
    float4 b = x0[t];
    x2[t] = make_float4(2.f * a.x - b.x, 2.f * a.y - b.y,
                        2.f * a.z - b.z, 2.f * a.w - b.w);
}

// ---------------- reparameterized weight/bias sample ----------------
__global__ __launch_bounds__(256) void sample_weights(const float* __restrict__ mu_w,
                                                      const float* __restrict__ ls_w,
                                                      const float* __restrict__ eps_w,
                                                      const float* __restrict__ mu_b,
                                                      const float* __restrict__ ls_b,
                                                      const float* __restrict__ eps_b,
                                                      float* __restrict__ w,
                                                      float* __restrict__ bias) {
    int t = blockIdx.x * 256 + threadIdx.x;
    if (t < DD * NUM_MAT * DOUT) w[t] = mu_w[t] + expf(ls_w[t]) * eps_w[t];
    if (t < DOUT)                bias[t] = mu_b[t] + expf(ls_b[t]) * eps_b[t];
}

// ---------------- WMMA GEMM pass: out (+)= X0 @ W_m0 [+ X1 @ W_m1] ----------------
// X layout [n][b][d]; output row r = b*N + n; K-dim = d (contiguous, 128).
// Block = 256 threads = 8 waves; block tile = 128 rows x 128 cols.
// Wave tile = 16 rows x 128 cols = 8 x v8f accumulators.
// Up to two pair-interleaved W panels staged in dynamic LDS (72 KB each).
__global__ __launch_bounds__(256) void gemm_wmma_pass(const float* __restrict__ X0, int m0,
                                                      const float* __restrict__ X1, int m1,
                                                      const float* __restrict__ Wall,
                                                      const float* __restrict__ bias,
                                                      float* __restrict__ out,
                                                      int init) {
    extern __shared__ float wlds[];
    const int npass = (X1 != nullptr) ? 2 : 1;

    for (int p = 0; p < npass; ++p) {
        int m = p ? m1 : m0;
        const float* Wp = Wall + m * DOUT;     // row f = d*NUM_MAT + m
        for (int q = threadIdx.x; q < DD * DOUT; q += 256) {
            int d = q >> 7, j = q & 127;
            wlds[p * PANEL + (d >> 1) * PSTRIDE + 2 * j + (d & 1)] =
                Wp[(size_t)d * (NUM_MAT * DOUT) + j];
        }
    }
    __syncthreads();

    const int r0   = blockIdx.x * 128;         // output row tile
    const int b    = r0 >> 12;                 // r = b*N + n,  N = 4096
    const int n0   = r0 & (NN - 1);
    const int wave = threadIdx.x >> 5;
    const int lane = threadIdx.x & 31;
    const int mrow = lane & 15;                // A: lanes 0-15 rows, 16-31 same rows
    const int koff = (lane >> 4) << 1;         // A: K = 0,1 (lo half) / 2,3 (hi half)
    const int chi  = (lane >> 4) << 3;         // C/D: hi lanes hold M+8
    const int rbase = r0 + wave * 16;
    const size_t arow_off = (size_t)(n0 + wave * 16 + mrow) * ROWLEN + b * DD;

    v8f acc[8];
    if (init) {
        #pragma unroll
        for (int nt = 0; nt < 8; ++nt) {
            float bv = bias[nt * 16 + mrow];   // C element depends on col only
            #pragma unroll
            for (int j = 0; j < 8; ++j) acc[nt][j] = bv;
        }
    } else {
        #pragma unroll
        for (int nt = 0; nt < 8; ++nt)
            #pragma unroll
            for (int j = 0; j < 8; ++j)
                acc[nt][j] = out[(size_t)(rbase + j + chi) * DOUT + nt * 16 + mrow];
    }

    for (int p = 0; p < npass; ++p) {
        const float* arow = (p ? X1 : X0) + arow_off;      // contiguous in d
        const float* wp = wlds + p * PANEL + 2 * mrow;
        for (int d0 = 0; d0 < DD; d0 += 4) {
            float2 av = *(const float2*)(arow + d0 + koff);   // 8B, aligned
            v2f a; a[0] = av.x; a[1] = av.y;
            const float* bbase = wp + ((d0 + koff) >> 1) * PSTRIDE;
            #pragma unroll
            for (int nt = 0; nt < 8; ++nt) {
                v2f bfr = *(const v2f*)(bbase + nt * 32);  // one ds_load_b64
                acc[nt] = __builtin_amdgcn_wmma_f32_16x16x4_f32(
                    false, a, false, bfr, (short)0, acc[nt], false, false);
            }
        }
    }

    #pragma unroll
    for (int nt = 0; nt < 8; ++nt)
        #pragma unroll
        for (int j = 0; j < 8; ++j)
            out[(size_t)(rbase + j + chi) * DOUT + nt * 16 + mrow] = acc[nt][j];
}

// ---------------- host-side orchestration ----------------
extern "C" void kernel_launch(void* const* d_in, const int* in_sizes, int n_in,
                              void* d_out, int out_size, void* d_ws, size_t ws_size,
                              hipStream_t stream) {
    (void)in_sizes; (void)n_in; (void)out_size; (void)ws_size;
    const float* inputs = (const float*)d_in[0];
    const float* state  = (const float*)d_in[1];
    const float* vals   = (const float*)d_in[2];
    const float* mu_w   = (const float*)d_in[3];
    const float* ls_w   = (const float*)d_in[4];
    const float* mu_b   = (const float*)d_in[5];
    const float* ls_b   = (const float*)d_in[6];
    const float* eps_w  = (const float*)d_in[7];
    const float* eps_b  = (const float*)d_in[8];
    const int*   rows   = (const int*)d_in[9];
    const int*   cols   = (const int*)d_in[10];
    float* out = (float*)d_out;

    // rolling workspace: 3 x 128MB diffusion buffers + sampled weights/bias
    float* buf0 = (float*)d_ws;
    float* buf1 = buf0 + XMAT_ELEMS;
    float* buf2 = buf1 + XMAT_ELEMS;
    float* wbuf = buf2 + XMAT_ELEMS;
    float* bbuf = wbuf + (DD * NUM_MAT * DOUT);

    const dim3 blk(256);
    const unsigned elemBlocks = (unsigned)(XMAT_ELEMS / 256);        // 131072
    const unsigned vec4Blocks = (unsigned)(XMAT_ELEMS / (256 * 4));  // 32768
    const unsigned gemmBlocks = (unsigned)((BB * NN) / 128);         // 2048
    const size_t sh1 = (size_t)PANEL * 4;                            // 72 KB
    const size_t sh2 = (size_t)2 * PANEL * 4;                        // 144 KB

    // xs0 = pack(inputs, state); sample weights
    pack_x0<<<elemBlocks, blk, 0, stream>>>(inputs, state, buf0);
    sample_weights<<<(DD * NUM_MAT * DOUT + 255) / 256, blk, 0, stream>>>(
        mu_w, ls_w, eps_w, mu_b, ls_b, eps_b, wbuf, bbuf);

    // xs1 = A0 @ xs0
    zero_buf<<<vec4Blocks, blk, 0, stream>>>((float4*)buf1);
    spmm_scatter<<<EE, blk, 0, stream>>>(rows + 0 * EE, cols + 0 * EE, vals + 0 * EE, buf0, buf1);

    // out = bias + xs0 @ W0 + xs1 @ W1
    gemm_wmma_pass<<<gemmBlocks, blk, sh2, stream>>>(buf0, 0, buf1, 1, wbuf, bbuf, out, 1);

    // xs2 = 2*(A0 @ xs1) - xs0
    zero_buf<<<vec4Blocks, blk, 0, stream>>>((float4*)buf2);
    spmm_scatter<<<EE, blk, 0, stream>>>(rows + 0 * EE, cols + 0 * EE, vals + 0 * EE, buf1, buf2);
    cheb_axpy<<<vec4Blocks, blk, 0, stream>>>((float4*)buf2, (const float4*)buf0);

    // xs3 = A1 @ xs2   (buf0 = xs0 dead -> reuse)
    zero_buf<<<vec4Blocks, blk, 0, stream>>>((float4*)buf0);
    spmm_scatter<<<EE, blk, 0, stream>>>(rows + 1 * EE, cols + 1 * EE, vals + 1 * EE, buf2, buf0);

    // out += xs2 @ W2 + xs3 @ W3
    gemm_wmma_pass<<<gemmBlocks, blk, sh2, stream>>>(buf2, 2, buf0, 3, wbuf, bbuf, out, 0);

    // xs4 = 2*(A1 @ xs3) - xs1   (buf2 = xs2 dead -> reuse)
    zero_buf<<<vec4Blocks, blk, 0, stream>>>((float4*)buf2);
    spmm_scatter<<<EE, blk, 0, stream>>>(rows + 1 * EE, cols + 1 * EE, vals + 1 * EE, buf0, buf2);
    cheb_axpy<<<vec4Blocks, blk, 0, stream>>>((float4*)buf2, (const float4*)buf1);

    // out += xs4 @ W4
    gemm_wmma_pass<<<gemmBlocks, blk, sh1, stream>>>(buf2, 4, nullptr, 0, wbuf, bbuf, out, 0);
}